// PointKernelPartialDense_50938312130931
// MI455X (gfx1250) — compile-verified
//
#include <hip/hip_runtime.h>

// Problem constants (from the reference)
#define N_PTS   131072
#define NEIGHB  32
#define KP      15
#define CIN     64
#define COUT    64
#define M_Q     32768
#define EXTENT  0.24f          // 1.2 * RADIUS / 5
#define BN_EPS  1e-5f
#define SLOPE   0.01f
#define WF_STRIDE 964          // 960 + 4 pad: stride mod 64 banks == 4 -> conflict-free

typedef __attribute__((ext_vector_type(2))) float v2f;
typedef __attribute__((ext_vector_type(8))) float v8f;

// ---------------------------------------------------------------------------
// Main kernel: gather -> nearest-kernel-point bucketing -> WMMA f32 GEMM
// Block = 256 threads (8 wave32), owns 32 queries.
// ---------------------------------------------------------------------------
__global__ __launch_bounds__(256) void kpconv_main(
    const float* __restrict__ x,        // [N, CIN]
    const int*   __restrict__ idx_nb,   // [N, NEIGHB]
    const float* __restrict__ pos_nb,   // [N, NEIGHB, 3]
    const int*   __restrict__ idx_s,    // [M]
    const float* __restrict__ kpts,     // [K, 3]
    const float* __restrict__ kw,       // [K*CIN, COUT] row-major
    float*       __restrict__ feats)    // [M, COUT]  (pre-BN, staged in d_out)
{
    __shared__ float wf[32 * WF_STRIDE];   // weighted_feats, 32 x (15*64) padded
    __shared__ float s_kp[48];
    __shared__ int   s_idx[32 * NEIGHB];
    __shared__ float s_w[32 * NEIGHB];
    __shared__ int   s_k[32 * NEIGHB];
    __shared__ int   s_q[32];

    const int tid = threadIdx.x;
    const int m0  = blockIdx.x * 32;

    if (tid < KP * 3) s_kp[tid] = kpts[tid];
    if (tid >= 64 && tid < 96) s_q[tid - 64] = idx_s[m0 + (tid - 64)];
    for (int i = tid; i < 32 * WF_STRIDE; i += 256) wf[i] = 0.f;
    __syncthreads();

    // ---- Stage 1: per (query m, neighbor n): nearest kernel point + weight
    for (int t = tid; t < 32 * NEIGHB; t += 256) {
        const int m = t >> 5, n = t & 31;
        const int q = s_q[m];
        const long pb = ((long)q * NEIGHB + n) * 3;
        const float px = pos_nb[pb], py = pos_nb[pb + 1], pz = pos_nb[pb + 2];
        float best = 3.4e38f; int bk = 0;
#pragma unroll
        for (int k = 0; k < KP; ++k) {
            const float dx = px - s_kp[3 * k + 0];
            const float dy = py - s_kp[3 * k + 1];
            const float dz = pz - s_kp[3 * k + 2];
            const float d = dx * dx + dy * dy + dz * dz;
            if (d < best) { best = d; bk = k; }   // first-min, matches argmin
        }
        const float w = fmaxf(1.f - sqrtf(best) * (1.f / EXTENT), 0.f);
        s_idx[t] = idx_nb[(long)q * NEIGHB + n];
        s_w[t]   = w;
        s_k[t]   = bk;
    }
    __syncthreads();

    // ---- Stage 2: bucket-accumulate w * x[neighbor] into wf[m][k*64 + c]
    // Thread owns (m, 4-channel group) exclusively -> no atomics needed.
    for (int t = tid; t < 32 * 16; t += 256) {
        const int m = t >> 4, cg = t & 15;
        float* row = &wf[m * WF_STRIDE + cg * 4];
        for (int n = 0; n < NEIGHB; ++n) {
            const int   nb = s_idx[m * NEIGHB + n];
            const float w  = s_w[m * NEIGHB + n];
            if (nb < N_PTS && w > 0.f) {
                const float4 xv =
                    *reinterpret_cast<const float4*>(x + (long)nb * CIN + cg * 4);
                float* dst = row + s_k[m * NEIGHB + n] * CIN;
                dst[0] += w * xv.x;
                dst[1] += w * xv.y;
                dst[2] += w * xv.z;
                dst[3] += w * xv.w;
            }
        }
    }
    __syncthreads();

    // ---- Stage 3: feats[16x16 tile] = wf[16 x 960] * kw[960 x 16] via WMMA f32
    const int wave  = tid >> 5, lane = tid & 31;
    const int mtile = wave >> 2;         // 0..1
    const int ntile = wave & 3;          // 0..3
    const int half  = lane >> 4;         // lanes 0-15 -> K pair {0,1}; 16-31 -> {2,3}
    const int l     = lane & 15;

    const float* Arow = &wf[(mtile * 16 + l) * WF_STRIDE];
    const float* Bcol = kw + ntile * 16 + l;

    v8f acc = {0.f, 0.f, 0.f, 0.f, 0.f, 0.f, 0.f, 0.f};
    for (int kf = 0; kf < KP * CIN; kf += 4) {
        const int kr = kf + 2 * half;
        v2f a; a.x = Arow[kr];                 // A[M=l][K=kr]
               a.y = Arow[kr + 1];             // A[M=l][K=kr+1]
        v2f b; b.x = Bcol[(long)kr * COUT];    // B[K=kr][N=l]
               b.y = Bcol[(long)(kr + 1) * COUT];
        acc = __builtin_amdgcn_wmma_f32_16x16x4_f32(
            false, a, false, b, (short)0, acc, false, false);
    }

    // D layout: VGPR j -> row (j + 8*half), col = l
    float* out = feats + (long)(m0 + mtile * 16) * COUT + ntile * 16 + l;
#pragma unroll
    for (int j = 0; j < 8; ++j)
        out[(long)(j + 8 * half) * COUT] = acc[j];
}

// ---------------------------------------------------------------------------
// BN batch statistics: one block per output channel
// ---------------------------------------------------------------------------
__global__ __launch_bounds__(256) void bn_stats(const float* __restrict__ feats,
                                                float* __restrict__ stats)
{
    __shared__ float rs[256], rs2[256];
    const int ch = blockIdx.x, tid = threadIdx.x;
    float s = 0.f, s2 = 0.f;
    for (int m = tid; m < M_Q; m += 256) {
        const float v = feats[(long)m * COUT + ch];
        s += v; s2 += v * v;
    }
    rs[tid] = s; rs2[tid] = s2;
    __syncthreads();
    for (int off = 128; off > 0; off >>= 1) {
        if (tid < off) { rs[tid] += rs[tid + off]; rs2[tid] += rs2[tid + off]; }
        __syncthreads();
    }
    if (tid == 0) {
        const float mean = rs[0] * (1.f / (float)M_Q);
        const float var  = rs2[0] * (1.f / (float)M_Q) - mean * mean;
        stats[ch]        = mean;
        stats[COUT + ch] = rsqrtf(var + BN_EPS);
    }
}

// ---------------------------------------------------------------------------
// Apply BN + LeakyReLU in place on d_out
// ---------------------------------------------------------------------------
__global__ __launch_bounds__(256) void bn_apply(float* __restrict__ feats,
                                                const float* __restrict__ stats,
                                                const float* __restrict__ gamma,
                                                const float* __restrict__ beta)
{
    const int i  = blockIdx.x * 256 + threadIdx.x;
    const int ch = i & (COUT - 1);
    const float v  = feats[i];
    const float bn = (v - stats[ch]) * stats[COUT + ch] * gamma[ch] + beta[ch];
    feats[i] = (bn >= 0.f) ? bn : SLOPE * bn;
}

extern "C" void kernel_launch(void* const* d_in, const int* in_sizes, int n_in,
                              void* d_out, int out_size, void* d_ws, size_t ws_size,
                              hipStream_t stream)
{
    (void)in_sizes; (void)n_in; (void)out_size; (void)ws_size;
    const float* x      = (const float*)d_in[0];
    const int*   idx_nb = (const int*)  d_in[1];
    const float* pos_nb = (const float*)d_in[2];
    const int*   idx_s  = (const int*)  d_in[3];
    const float* kpts   = (const float*)d_in[4];
    const float* kw     = (const float*)d_in[5];
    const float* gamma  = (const float*)d_in[6];
    const float* beta   = (const float*)d_in[7];
    float* feats = (float*)d_out;     // pre-BN activations staged in d_out
    float* stats = (float*)d_ws;      // 128 floats: mean[64], rstd[64]

    kpconv_main<<<M_Q / 32, 256, 0, stream>>>(x, idx_nb, pos_nb, idx_s, kpts, kw, feats);
    bn_stats  <<<COUT, 256, 0, stream>>>(feats, stats);
    bn_apply  <<<(M_Q * COUT) / 256, 256, 0, stream>>>(feats, stats, gamma, beta);
}